// DifferentiableTopK_29480655520109
// MI455X (gfx1250) — compile-verified
//
#include <hip/hip_runtime.h>

typedef __attribute__((ext_vector_type(2))) float v2f;
typedef __attribute__((ext_vector_type(8))) float v8f;

#define NN 512
#define KTOP 50
#define NITERS 100
#define INVEPS 1000.0f   // 1/EPSILON, EPSILON = 1e-3

// One workgroup per batch row. 512 threads = 16 wave32.
// LDS state: sf[i] = (s_i, f_i), ssg[j] = (sorted_s_j, g_j).  log_P stays
// implicit: log_P[i,j] = -INVEPS*(s_i - ss_j)^2 + f_i + g_j.
__global__ __launch_bounds__(NN, 1)
void sinkhorn_topk(const float* __restrict__ scores, float* __restrict__ out) {
  __shared__ float2 sf[NN];
  __shared__ float2 ssg[NN];

  const int tid = threadIdx.x;
  const int b   = blockIdx.x;

  const float sval = scores[b * NN + tid];
  sf[tid]  = make_float2(sval, 0.0f);   // f = 0
  ssg[tid] = make_float2(sval, 0.0f);   // g = 0; .x sorted below
  __syncthreads();

  // ---- bitonic sort, descending, on ssg[].x ----
  for (int k = 2; k <= NN; k <<= 1) {
    for (int j = k >> 1; j > 0; j >>= 1) {
      const int ixj = tid ^ j;
      if (ixj > tid) {
        const float a = ssg[tid].x;
        const float c = ssg[ixj].x;
        const bool up = ((tid & k) == 0);
        if (up ? (a < c) : (a > c)) {
          ssg[tid].x = c;
          ssg[ixj].x = a;
        }
      }
      __syncthreads();
    }
  }

  const float ssj = ssg[tid].x;   // this thread's column key
  const float si  = sf[tid].x;    // this thread's row key

  for (int it = 0; it < NITERS; ++it) {
    // ---- column normalize: g[j] = -LSE_i( -1000*(s_i-ss_j)^2 + f_i ) ----
    float m = -INFINITY;
#pragma unroll 8
    for (int i = 0; i < NN; ++i) {
      const float2 p = sf[i];
      const float d = p.x - ssj;
      const float t = __builtin_fmaf(d * -INVEPS, d, p.y);
      m = fmaxf(m, t);
    }
    float sum = 0.0f;
#pragma unroll 8
    for (int i = 0; i < NN; ++i) {
      const float2 p = sf[i];
      const float d = p.x - ssj;
      const float t = __builtin_fmaf(d * -INVEPS, d, p.y);
      sum += __expf(t - m);
    }
    ssg[tid].y = -(m + __logf(sum));
    __syncthreads();

    // ---- row normalize: f[i] = -LSE_j( -1000*(s_i-ss_j)^2 + g_j ) ----
    m = -INFINITY;
#pragma unroll 8
    for (int j = 0; j < NN; ++j) {
      const float2 q = ssg[j];
      const float d = si - q.x;
      const float t = __builtin_fmaf(d * -INVEPS, d, q.y);
      m = fmaxf(m, t);
    }
    sum = 0.0f;
#pragma unroll 8
    for (int j = 0; j < NN; ++j) {
      const float2 q = ssg[j];
      const float d = si - q.x;
      const float t = __builtin_fmaf(d * -INVEPS, d, q.y);
      sum += __expf(t - m);
    }
    sf[tid].y = -(m + __logf(sum));
    __syncthreads();
  }

  // ---- finalize: out[i] = sum_{j<K} exp(logP0 + f_i + g_j) via WMMA ----
  // V_WMMA_F32_16X16X4_F32 row-sum: A = 16x4 tile of P (A layout: lane L
  // holds row M=L%16; VGPR0/1 hold K = {0,1} for lanes 0-15, {2,3} for
  // lanes 16-31), B = ones(4x16) (layout-invariant), D accumulates exact
  // f32 row sums replicated across columns.
  const int lane = tid & 31;
  const int wave = tid >> 5;
  const int lm   = lane & 15;
  const int half = lane >> 4;      // 0: K={0,1}, 1: K={2,3}

  v2f bones; bones[0] = 1.0f; bones[1] = 1.0f;

  for (int gsel = 0; gsel < 2; ++gsel) {
    const int r0 = wave * 32 + gsel * 16;   // 16-row group base
    const float2 p = sf[r0 + lm];
    const float si2 = p.x;
    const float fi2 = p.y;
    const int jk = half * 2;
    v8f acc = {};
    for (int j0 = 0; j0 < KTOP; j0 += 4) {  // j0 = 0,4,...,48 (13 WMMAs)
      v2f a;
#pragma unroll
      for (int u = 0; u < 2; ++u) {
        const int j = j0 + jk + u;
        const float2 q = ssg[j];
        const float d = si2 - q.x;
        const float t = __builtin_fmaf(d * -INVEPS, d, fi2 + q.y);
        a[u] = (j < KTOP) ? __expf(t) : 0.0f;  // mask j = 50..51
      }
      acc = __builtin_amdgcn_wmma_f32_16x16x4_f32(
          false, a, false, bones, (short)0, acc, false, false);
    }
    // D layout: lanes 0-15 hold M=0..7 in acc[0..7]; lanes 16-31 hold M=8..15.
    if (lm == 0) {
      float* o = out + b * NN + r0 + half * 8;
#pragma unroll
      for (int r = 0; r < 8; ++r) o[r] = acc[r];
    }
  }
}

extern "C" void kernel_launch(void* const* d_in, const int* in_sizes, int n_in,
                              void* d_out, int out_size, void* d_ws, size_t ws_size,
                              hipStream_t stream) {
  const float* scores = (const float*)d_in[0];
  float* out = (float*)d_out;
  const int B = in_sizes[0] / NN;   // 16 for the reference shapes
  sinkhorn_topk<<<dim3(B), dim3(NN), 0, stream>>>(scores, out);
}